// Smpl_28424093565597
// MI455X (gfx1250) — compile-verified
//
#include <hip/hip_runtime.h>
#include <math.h>

#define BB 512
#define VV 6890
#define NJ 24
#define NJO 19
#define NC (VV*3)      // 20670
#define NPF 207
#define PFS 208        // pose_feature padded stride (col 207 zeroed)

typedef __attribute__((ext_vector_type(2))) float v2f;
typedef __attribute__((ext_vector_type(8))) float v8f;

// ---------------------------------------------------------------------------
// K1: Rodrigues. One thread per (b, joint). Writes rot to d_out rot region and
// pose_feature (rot[1:]-I flattened, B x 208 padded, col 207 = 0) to ws.
// ---------------------------------------------------------------------------
__global__ void k_rodrigues(const float* __restrict__ inputs,
                            float* __restrict__ rot_out,
                            float* __restrict__ pf) {
    int idx = blockIdx.x * blockDim.x + threadIdx.x;
    if (idx >= BB * NJ) return;
    int b = idx / NJ, j = idx % NJ;
    float r0 = inputs[b * 82 + 3 * j + 0];
    float r1 = inputs[b * 82 + 3 * j + 1];
    float r2 = inputs[b * 82 + 3 * j + 2];
    float a0 = r0 + 1e-8f, a1 = r1 + 1e-8f, a2 = r2 + 1e-8f;
    float ang = sqrtf(a0 * a0 + a1 * a1 + a2 * a2);
    float inv = 1.0f / ang;
    float n0 = r0 * inv, n1 = r1 * inv, n2 = r2 * inv;
    float c = cosf(ang), s = sinf(ang), oc = 1.0f - c;
    float R[9];
    R[0] = c + oc * n0 * n0;  R[1] = oc * n0 * n1 - s * n2; R[2] = oc * n0 * n2 + s * n1;
    R[3] = oc * n1 * n0 + s * n2; R[4] = c + oc * n1 * n1;  R[5] = oc * n1 * n2 - s * n0;
    R[6] = oc * n2 * n0 - s * n1; R[7] = oc * n2 * n1 + s * n0; R[8] = c + oc * n2 * n2;
    float* ro = rot_out + b * (NJ * 9) + j * 9;
#pragma unroll
    for (int q = 0; q < 9; ++q) ro[q] = R[q];
    if (j >= 1) {
        float* p = pf + b * PFS + (j - 1) * 9;
#pragma unroll
        for (int q = 0; q < 9; ++q)
            p[q] = R[q] - ((q == 0 || q == 4 || q == 8) ? 1.0f : 0.0f);
    } else {
        pf[b * PFS + NPF] = 0.0f;   // zero the pad column
    }
}

// ---------------------------------------------------------------------------
// K2: v_shaped = beta(512x10) @ shapes(10x20670) + v_template via fp32 WMMA
// 16x16x4. One wave32 per 16x64 strip (4 accumulators share each A frag).
// All loads branchless: clamped addresses + cndmask-zero on the K pad only.
// ---------------------------------------------------------------------------
__global__ __launch_bounds__(32) void k_vshaped(const float* __restrict__ inputs,
                                                const float* __restrict__ shapes,
                                                const float* __restrict__ v_template,
                                                float* __restrict__ vbuf) {
    const int lane  = threadIdx.x & 31;
    const int lan   = lane & 15;
    const int khalf = lane >> 4;
    const int nb = blockIdx.x * 64;
    const int m0 = blockIdx.y * 16;
    const int m  = m0 + lan;

    int  nidx[4];
    bool nok[4];
#pragma unroll
    for (int t = 0; t < 4; ++t) {
        int n = nb + t * 16 + lan;
        nok[t]  = (n < NC);
        nidx[t] = min(n, NC - 1);
    }

    v8f acc[4] = {};
#pragma unroll
    for (int k0 = 0; k0 < 12; k0 += 4) {   // K=10 padded to 12
        int ka = k0 + 2 * khalf;
        int c0 = min(ka, 9), c1 = min(ka + 1, 9);
        float ax = inputs[m * 82 + 72 + c0];     // unconditional clamped load
        float ay = inputs[m * 82 + 72 + c1];
        v2f a;
        a.x = (ka     < 10) ? ax : 0.0f;         // cndmask, no branch
        a.y = (ka + 1 < 10) ? ay : 0.0f;
#pragma unroll
        for (int t = 0; t < 4; ++t) {
            v2f b;                                // B clamped-finite; A zero cancels pad
            b.x = shapes[c0 * NC + nidx[t]];
            b.y = shapes[c1 * NC + nidx[t]];
            acc[t] = __builtin_amdgcn_wmma_f32_16x16x4_f32(false, a, false, b,
                                                           (short)0, acc[t], false, false);
        }
    }
#pragma unroll
    for (int t = 0; t < 4; ++t) {
        if (nok[t]) {
            float tv = v_template[nidx[t]];
#pragma unroll
            for (int r = 0; r < 8; ++r)
                vbuf[(m0 + r + 8 * khalf) * NC + nidx[t]] = acc[t][r] + tv;
        }
    }
}

// ---------------------------------------------------------------------------
// K4: vbuf += pose_feature(512x207) @ posedirs(207x20670), fp32 WMMA 16x16x4,
// accumulating in place through the C operand (vbuf already holds v_shaped).
// Main K loop (k<204) is completely guard-free: A frags are contiguous b64
// loads from the padded pf; the single tail step clamps B rows (A pad = 0).
// ---------------------------------------------------------------------------
__global__ __launch_bounds__(32) void k_vposed(const float* __restrict__ pf,
                                               const float* __restrict__ posedirs,
                                               float* __restrict__ vbuf) {
    const int lane  = threadIdx.x & 31;
    const int lan   = lane & 15;
    const int khalf = lane >> 4;
    const int nb = blockIdx.x * 64;
    const int m0 = blockIdx.y * 16;
    const int m  = m0 + lan;

    int  nidx[4];
    bool nok[4];
#pragma unroll
    for (int t = 0; t < 4; ++t) {
        int n = nb + t * 16 + lan;
        nok[t]  = (n < NC);
        nidx[t] = min(n, NC - 1);
    }

    v8f acc[4];
#pragma unroll
    for (int t = 0; t < 4; ++t)
#pragma unroll
        for (int r = 0; r < 8; ++r)
            acc[t][r] = vbuf[(m0 + r + 8 * khalf) * NC + nidx[t]];

    const float* pfm = pf + m * PFS;

#pragma unroll 2
    for (int k0 = 0; k0 < 204; k0 += 4) {    // guard-free main loop
        int ka = k0 + 2 * khalf;
        v2f a;
        a.x = pfm[ka];
        a.y = pfm[ka + 1];
        __builtin_prefetch(&posedirs[min(k0 + 8, NPF - 1) * NC + nidx[0]], 0, 1);
#pragma unroll
        for (int t = 0; t < 4; ++t) {
            v2f b;
            b.x = posedirs[ka * NC + nidx[t]];
            b.y = posedirs[(ka + 1) * NC + nidx[t]];
            acc[t] = __builtin_amdgcn_wmma_f32_16x16x4_f32(false, a, false, b,
                                                           (short)0, acc[t], false, false);
        }
    }
    {   // tail: k = 204..207; pf pad column 207 is exactly 0, B row clamped
        int ka = 204 + 2 * khalf;
        int r0 = min(ka, NPF - 1), r1 = min(ka + 1, NPF - 1);
        v2f a;
        a.x = pfm[ka];
        a.y = pfm[ka + 1];
#pragma unroll
        for (int t = 0; t < 4; ++t) {
            v2f b;
            b.x = posedirs[r0 * NC + nidx[t]];
            b.y = posedirs[r1 * NC + nidx[t]];
            acc[t] = __builtin_amdgcn_wmma_f32_16x16x4_f32(false, a, false, b,
                                                           (short)0, acc[t], false, false);
        }
    }
#pragma unroll
    for (int t = 0; t < 4; ++t) {
        if (nok[t]) {
#pragma unroll
            for (int r = 0; r < 8; ++r)
                vbuf[(m0 + r + 8 * khalf) * NC + nidx[t]] = acc[t][r];
        }
    }
}

// ---------------------------------------------------------------------------
// K3/K7: joint regressions. One block per batch; thread t owns output (c,j),
// full reduction over V. Regressor stays resident in the 192 MB L2.
// ---------------------------------------------------------------------------
__global__ __launch_bounds__(96) void k_vjoints(const float* __restrict__ vbuf,
                                                const float* __restrict__ reg,
                                                float* __restrict__ Jv) {
    int b = blockIdx.x, t = threadIdx.x;
    if (t >= NJ * 3) return;
    int j = t % NJ, c = t / NJ;
    const float* vb = vbuf + b * NC;
    float s = 0.0f;
    for (int v = 0; v < VV; ++v)
        s += vb[v * 3 + c] * reg[v * NJ + j];
    Jv[b * (NJ * 3) + j * 3 + c] = s;
}

__global__ __launch_bounds__(64) void k_joints(const float* __restrict__ vbuf,
                                               const float* __restrict__ reg,
                                               float* __restrict__ jout) {
    int b = blockIdx.x, t = threadIdx.x;
    if (t >= NJO * 3) return;
    int j = t % NJO, c = t / NJO;
    const float* vb = vbuf + b * NC;
    float s = 0.0f;
    for (int v = 0; v < VV; ++v)
        s += vb[v * 3 + c] * reg[v * NJO + j];
    jout[b * (NJO * 3) + j * 3 + c] = s;
}

// ---------------------------------------------------------------------------
// K5: kinematic chain. One thread per batch; ancestor affine transforms
// (3x4 each) kept in LDS (32 threads * 24 * 12 floats = 36 KB << 320 KB WGP).
// Emits A' = A_world with translation column replaced by t_w - R_w * J_i.
// ---------------------------------------------------------------------------
__global__ __launch_bounds__(32) void k_rigid(const float* __restrict__ rot,
                                              const float* __restrict__ Jv,
                                              const int* __restrict__ parents,
                                              float* __restrict__ Aout) {
    __shared__ float chain[32][NJ][12];
    int tid = threadIdx.x;
    int b = blockIdx.x * 32 + tid;
    const float* Rb = rot + b * (NJ * 9);
    const float* Jb = Jv + b * (NJ * 3);

    for (int i = 0; i < NJ; ++i) {
        float R[9];
#pragma unroll
        for (int q = 0; q < 9; ++q) R[q] = Rb[i * 9 + q];
        float* A = chain[tid][i];
        if (i == 0) {
            A[0] = R[0]; A[1] = R[1]; A[2]  = R[2]; A[3]  = Jb[0];
            A[4] = R[3]; A[5] = R[4]; A[6]  = R[5]; A[7]  = Jb[1];
            A[8] = R[6]; A[9] = R[7]; A[10] = R[8]; A[11] = Jb[2];
        } else {
            int p = parents[i];
            float t[3];
#pragma unroll
            for (int c = 0; c < 3; ++c) t[c] = Jb[i * 3 + c] - Jb[p * 3 + c];
            const float* P = chain[tid][p];
#pragma unroll
            for (int r = 0; r < 3; ++r) {
#pragma unroll
                for (int c = 0; c < 3; ++c)
                    A[r * 4 + c] = P[r * 4 + 0] * R[0 * 3 + c]
                                 + P[r * 4 + 1] * R[1 * 3 + c]
                                 + P[r * 4 + 2] * R[2 * 3 + c];
                A[r * 4 + 3] = P[r * 4 + 0] * t[0] + P[r * 4 + 1] * t[1]
                             + P[r * 4 + 2] * t[2] + P[r * 4 + 3];
            }
        }
    }
    float* outb = Aout + b * (NJ * 16);
    for (int i = 0; i < NJ; ++i) {
        const float* A = chain[tid][i];
        float jx = Jb[i * 3 + 0], jy = Jb[i * 3 + 1], jz = Jb[i * 3 + 2];
#pragma unroll
        for (int r = 0; r < 3; ++r) {
            float tb = A[r * 4 + 3] - (A[r * 4 + 0] * jx + A[r * 4 + 1] * jy + A[r * 4 + 2] * jz);
            outb[i * 16 + r * 4 + 0] = A[r * 4 + 0];
            outb[i * 16 + r * 4 + 1] = A[r * 4 + 1];
            outb[i * 16 + r * 4 + 2] = A[r * 4 + 2];
            outb[i * 16 + r * 4 + 3] = tb;
        }
        outb[i * 16 + 12] = 0.0f; outb[i * 16 + 13] = 0.0f;
        outb[i * 16 + 14] = 0.0f; outb[i * 16 + 15] = 1.0f;
    }
}

// ---------------------------------------------------------------------------
// K6: fused skinning (never materializes T, saving 225 MB of traffic).
// A'[b] (1.5 KB) staged in LDS; vbuf updated in place (thread-private RMW).
// ---------------------------------------------------------------------------
__global__ __launch_bounds__(256) void k_skin(const float* __restrict__ Ap,
                                              const float* __restrict__ lbs,
                                              float* __restrict__ vbuf) {
    __shared__ float As[NJ * 16];
    int b = blockIdx.y, t = threadIdx.x;
    for (int i = t; i < NJ * 16; i += 256) As[i] = Ap[b * (NJ * 16) + i];
    __syncthreads();
    int v = blockIdx.x * 256 + t;
    if (v >= VV) return;
    float x = vbuf[b * NC + v * 3 + 0];
    float y = vbuf[b * NC + v * 3 + 1];
    float z = vbuf[b * NC + v * 3 + 2];
    float ax = 0.0f, ay = 0.0f, az = 0.0f;
#pragma unroll 4
    for (int j = 0; j < NJ; ++j) {
        float w = lbs[v * NJ + j];
        const float* A = &As[j * 16];
        ax += w * (A[0] * x + A[1] * y + A[2]  * z + A[3]);
        ay += w * (A[4] * x + A[5] * y + A[6]  * z + A[7]);
        az += w * (A[8] * x + A[9] * y + A[10] * z + A[11]);
    }
    vbuf[b * NC + v * 3 + 0] = ax;
    vbuf[b * NC + v * 3 + 1] = ay;
    vbuf[b * NC + v * 3 + 2] = az;
}

extern "C" void kernel_launch(void* const* d_in, const int* in_sizes, int n_in,
                              void* d_out, int out_size, void* d_ws, size_t ws_size,
                              hipStream_t stream) {
    const float* inputs     = (const float*)d_in[0];
    const float* v_template = (const float*)d_in[1];
    const float* shapes     = (const float*)d_in[2];
    const float* posedirs   = (const float*)d_in[3];
    const float* smpl_reg   = (const float*)d_in[4];
    const float* lbs        = (const float*)d_in[5];
    const float* joint_reg  = (const float*)d_in[6];
    const int*   parents    = (const int*)d_in[7];

    float* out       = (float*)d_out;
    float* verts     = out;                                  // 512*20670
    float* joints_o  = out + (size_t)BB * NC;                // 512*57
    float* rot_out   = out + (size_t)BB * NC + BB * NJO * 3; // 512*216

    float* ws = (float*)d_ws;
    float* pf = ws;                          // 512*208 = 106496 floats (padded)
    float* Jv = ws + BB * PFS;               // 512*72  = 36864
    float* Ap = ws + BB * PFS + BB * NJ * 3; // 512*384 = 196608

    k_rodrigues<<<(BB * NJ + 255) / 256, 256, 0, stream>>>(inputs, rot_out, pf);

    dim3 gGemm((NC + 63) / 64, BB / 16);     // 323 x 32 strips of 16x64
    k_vshaped<<<gGemm, 32, 0, stream>>>(inputs, shapes, v_template, verts);
    k_vjoints<<<BB, 96, 0, stream>>>(verts, smpl_reg, Jv);
    k_vposed<<<gGemm, 32, 0, stream>>>(pf, posedirs, verts);
    k_rigid<<<BB / 32, 32, 0, stream>>>(rot_out, Jv, parents, Ap);

    dim3 gSkin((VV + 255) / 256, BB);
    k_skin<<<gSkin, 256, 0, stream>>>(Ap, lbs, verts);
    k_joints<<<BB, 64, 0, stream>>>(verts, joint_reg, joints_o);
}